// RvNN_71829033058692
// MI455X (gfx1250) — compile-verified
//
#include <hip/hip_runtime.h>
#include <hip/hip_bf16.h>

typedef __attribute__((ext_vector_type(2))) float v2f;
typedef __attribute__((ext_vector_type(8))) float v8f;

#define NUM_LEAVES_C 4096
#define NUM_PARENTS_C 4096
#define NUM_NODES_C 8192
#define L_C 512
#define H_C 5

__device__ __forceinline__ float hsig(float x) {
    return fminf(fmaxf(0.2f * x + 0.5f, 0.0f), 1.0f);
}

// Read one 8-byte-aligned pair (K=k0, K=k0+1) from a zero-padded [16][8] LDS tile.
// base points at row 0; m is the row (M for A, N for B); k0 = kc*4 + 2*g is even.
__device__ __forceinline__ v2f frag_pair(const float* base, int m, int g, int kc) {
    const int k0 = kc * 4 + 2 * g;                       // 0,2,4,6 -> 8B aligned
    const float2 t = *(const float2*)(base + m * 8 + k0); // ds_load_b64
    v2f r; r[0] = t.x; r[1] = t.y;
    return r;
}

// 5x5 matvec / mat-multicol via two K-chunks of V_WMMA_F32_16X16X4_F32.
// wbase: zero-padded 16x8 weight tile; b0/b1: B fragments for K-chunks 0/1.
__device__ __forceinline__ v8f wmma5(const float* wbase, v2f b0, v2f b1, int n16, int g) {
    v8f acc = {};
    v2f a0 = frag_pair(wbase, n16, g, 0);
    v2f a1 = frag_pair(wbase, n16, g, 1);
    acc = __builtin_amdgcn_wmma_f32_16x16x4_f32(false, a0, false, b0, (short)0, acc, false, false);
    acc = __builtin_amdgcn_wmma_f32_16x16x4_f32(false, a1, false, b1, (short)0, acc, false, false);
    return acc;
}

__global__ __launch_bounds__(32) void rvnn_last_h_kernel(
    const float* __restrict__ x_word, const int* __restrict__ x_index,
    const int* __restrict__ tree, const float* __restrict__ E,
    const float* __restrict__ W_z, const float* __restrict__ U_z,
    const float* __restrict__ W_r, const float* __restrict__ U_r,
    const float* __restrict__ W_h, const float* __restrict__ U_h,
    const float* __restrict__ b_z, const float* __restrict__ b_r,
    const float* __restrict__ b_h, float* __restrict__ out)
{
    const int lane = threadIdx.x;       // 0..31, one wave32
    const int n16  = lane & 15;
    const int g    = lane >> 4;

    // Fully zero-padded operand tiles -> unconditional ds_load_b64 fragments.
    __shared__ __align__(16) float s_W[6][16][8];   // 0:W_z 1:W_h 2:W_r 3:U_z 4:U_r 5:U_h
    __shared__ __align__(16) float s_xe[16][8];     // B cols: 0=xe_c0, 1=xe_c1, 2=xe_p
    __shared__ __align__(16) float s_bv[2][16][8];  // single-col B: 0=ht, 1=ht*r (row 0)
    __shared__ float s_az[3][8], s_ah[3][8], s_ar[8];
    __shared__ float s_uz[8], s_ur[8], s_uh[8];

    // ---- zero-init padded LDS (6*128 + 128 + 2*128 = 1152 floats) ----
    {
        float* p = (float*)s_W;   // s_W, s_xe, s_bv are contiguous enough to do separately
        for (int i = lane; i < 6 * 16 * 8; i += 32) p[i] = 0.0f;
        float* q = (float*)s_xe;
        for (int i = lane; i < 16 * 8; i += 32) q[i] = 0.0f;
        float* r = (float*)s_bv;
        for (int i = lane; i < 2 * 16 * 8; i += 32) r[i] = 0.0f;
    }
    __syncthreads();

    // ---- load 5x5 weights into padded tiles ----
    {
        const float* Ws[6] = { W_z, W_h, W_r, U_z, U_r, U_h };
        for (int w = 0; w < 6; ++w)
            if (lane < 25) s_W[w][lane / 5][lane % 5] = Ws[w][lane];
    }

    // ---- only the last scan step reaches the output; children are leaves ----
    const int c0 = tree[(NUM_PARENTS_C - 1) * 3 + 0];
    const int c1 = tree[(NUM_PARENTS_C - 1) * 3 + 1];
    const float e0 = (c0 > -1) ? 1.0f : 0.0f;
    const float e1 = (c1 > -1) ? 1.0f : 0.0f;
    const int n0 = min(max(c0, 0), NUM_NODES_C - 1);
    const int n1 = min(max(c1, 0), NUM_NODES_C - 1);
    const int nodes[3] = { n0, n1, NUM_NODES_C - 1 };

    // ---- phase 1: xe[n,h] = sum_l x_word[n,l] * E[x_index[n,l], h] for 3 nodes ----
    // Each lane owns 16 consecutive l -> float4/int4 streaming loads; E gathered scalar.
    for (int s = 0; s < 3; ++s) {
        const int base = nodes[s] * L_C + lane * (L_C / 32);
        const float4* wp = (const float4*)(x_word + base);
        const int4*   ip = (const int4*)(x_index + base);
        float a0 = 0.f, a1 = 0.f, a2 = 0.f, a3 = 0.f, a4 = 0.f;
        #pragma unroll
        for (int q = 0; q < 4; ++q) {                 // 4 * vec4 = 16 elements
            const float4 w4 = wp[q];
            const int4   i4 = ip[q];
            const float wv[4] = { w4.x, w4.y, w4.z, w4.w };
            const int    iv[4] = { i4.x, i4.y, i4.z, i4.w };
            #pragma unroll
            for (int j = 0; j < 4; ++j) {
                const float w = wv[j];
                const int   e = iv[j] * H_C;
                a0 = fmaf(w, E[e + 0], a0);
                a1 = fmaf(w, E[e + 1], a1);
                a2 = fmaf(w, E[e + 2], a2);
                a3 = fmaf(w, E[e + 3], a3);
                a4 = fmaf(w, E[e + 4], a4);
            }
        }
        for (int off = 16; off > 0; off >>= 1) {
            a0 += __shfl_down(a0, off, 32);
            a1 += __shfl_down(a1, off, 32);
            a2 += __shfl_down(a2, off, 32);
            a3 += __shfl_down(a3, off, 32);
            a4 += __shfl_down(a4, off, 32);
        }
        if (lane == 0) {
            s_xe[s][0] = a0; s_xe[s][1] = a1; s_xe[s][2] = a2;
            s_xe[s][3] = a3; s_xe[s][4] = a4;
        }
    }
    __syncthreads();

    // ---- round 1: [W_z|W_h|W_r] @ [xe0|xe1|xe2]  (6 WMMAs, shared B) ----
    v2f bA0 = frag_pair(&s_xe[0][0], n16, g, 0);
    v2f bA1 = frag_pair(&s_xe[0][0], n16, g, 1);
    v8f az = wmma5(&s_W[0][0][0], bA0, bA1, n16, g);
    v8f ah = wmma5(&s_W[1][0][0], bA0, bA1, n16, g);
    v8f ar = wmma5(&s_W[2][0][0], bA0, bA1, n16, g);
    if (g == 0 && n16 < 3) {
        #pragma unroll
        for (int h = 0; h < H_C; ++h) { s_az[n16][h] = az[h]; s_ah[n16][h] = ah[h]; }
        if (n16 == 2) {
            #pragma unroll
            for (int h = 0; h < H_C; ++h) s_ar[h] = ar[h];
        }
    }
    __syncthreads();

    // ---- leaf GRU + ht ----
    float z_l = 0.0f, ht_l = 0.0f;
    if (lane < H_C) {
        const int h = lane;
        const float z0  = hsig(s_az[0][h] + b_z[h]);
        const float cc0 = tanhf(s_ah[0][h] + b_h[h]);
        const float lh0 = (1.0f - z0) * cc0 * e0;
        const float z1  = hsig(s_az[1][h] + b_z[h]);
        const float cc1 = tanhf(s_ah[1][h] + b_h[h]);
        const float lh1 = (1.0f - z1) * cc1 * e1;
        ht_l = lh0 + lh1;
        s_bv[0][0][h] = ht_l;
    }
    __syncthreads();

    // ---- round 2: U_z @ ht, U_r @ ht  (4 WMMAs) ----
    v2f bB0 = frag_pair(&s_bv[0][0][0], n16, g, 0);
    v2f bB1 = frag_pair(&s_bv[0][0][0], n16, g, 1);
    v8f auz = wmma5(&s_W[3][0][0], bB0, bB1, n16, g);
    v8f aur = wmma5(&s_W[4][0][0], bB0, bB1, n16, g);
    if (lane == 0) {
        #pragma unroll
        for (int h = 0; h < H_C; ++h) { s_uz[h] = auz[h]; s_ur[h] = aur[h]; }
    }
    __syncthreads();

    if (lane < H_C) {
        const int h = lane;
        z_l = hsig(s_az[2][h] + s_uz[h] + b_z[h]);
        const float r_l = hsig(s_ar[h] + s_ur[h] + b_r[h]);
        s_bv[1][0][h] = ht_l * r_l;
    }
    __syncthreads();

    // ---- round 3: U_h @ (ht * r)  (2 WMMAs) ----
    v2f bC0 = frag_pair(&s_bv[1][0][0], n16, g, 0);
    v2f bC1 = frag_pair(&s_bv[1][0][0], n16, g, 1);
    v8f auh = wmma5(&s_W[5][0][0], bC0, bC1, n16, g);
    if (lane == 0) {
        #pragma unroll
        for (int h = 0; h < H_C; ++h) s_uh[h] = auh[h];
    }
    __syncthreads();

    // ---- final h = z*ht + (1-z)*tanh(W_h@xe_p + U_h@(ht*r) + b_h) ----
    if (lane < H_C) {
        const int h = lane;
        const float cc = tanhf(s_ah[2][h] + s_uh[h] + b_h[h]);
        out[h] = z_l * ht_l + (1.0f - z_l) * cc;
    }
}

extern "C" void kernel_launch(void* const* d_in, const int* in_sizes, int n_in,
                              void* d_out, int out_size, void* d_ws, size_t ws_size,
                              hipStream_t stream) {
    (void)in_sizes; (void)n_in; (void)out_size; (void)d_ws; (void)ws_size;
    const float* x_word  = (const float*)d_in[0];
    const int*   x_index = (const int*)  d_in[1];
    const int*   tree    = (const int*)  d_in[2];
    const float* E       = (const float*)d_in[3];
    const float* W_z     = (const float*)d_in[4];
    const float* U_z     = (const float*)d_in[5];
    const float* W_r     = (const float*)d_in[6];
    const float* U_r     = (const float*)d_in[7];
    const float* W_h     = (const float*)d_in[8];
    const float* U_h     = (const float*)d_in[9];
    const float* b_z     = (const float*)d_in[10];
    const float* b_r     = (const float*)d_in[11];
    const float* b_h     = (const float*)d_in[12];
    float* out = (float*)d_out;

    rvnn_last_h_kernel<<<dim3(1), dim3(32), 0, stream>>>(
        x_word, x_index, tree, E, W_z, U_z, W_r, U_r, W_h, U_h, b_z, b_r, b_h, out);
}